// OracleNetGNN_80900003987703
// MI455X (gfx1250) — compile-verified
//
#include <hip/hip_runtime.h>
#include <hip/hip_bf16.h>

// ---------------------------------------------------------------------------
// OracleNetGNN for MI455X (gfx1250): 2x GCNConv(128) + ReLU + mean-pool + linear
// fp32 throughout; GEMMs use V_WMMA_F32_16X16X4_F32 (wave32 WMMA).
// ---------------------------------------------------------------------------

typedef float v2f __attribute__((ext_vector_type(2)));
typedef float v8f __attribute__((ext_vector_type(8)));

#define F_DIM 128
#define LDS_PITCH 132   // 16 rows x 132 floats: (4*m + k) % 64 distinct per lane -> conflict-free

// -------------------- generic fill --------------------
__global__ void gnn_fill_f32(float* __restrict__ p, float v, long long n) {
  long long i = (long long)blockIdx.x * blockDim.x + threadIdx.x;
  long long stride = (long long)gridDim.x * blockDim.x;
  for (; i < n; i += stride) p[i] = v;
}

// -------------------- degree accumulation --------------------
__global__ void gnn_deg_accum(const int* __restrict__ ei, const float* __restrict__ ew,
                              float* __restrict__ deg, int nE) {
  int e = blockIdx.x * blockDim.x + threadIdx.x;
  if (e >= nE) return;
  int c = ei[(long long)nE + e];  // col row of edge_index
  atomicAdd(&deg[c], ew[e]);
}

__global__ void gnn_rsqrt_inplace(float* __restrict__ d, int n) {
  int i = blockIdx.x * blockDim.x + threadIdx.x;
  if (i >= n) return;
  float v = d[i];
  d[i] = (v > 0.0f) ? rsqrtf(v) : 0.0f;
}

// -------------------- WMMA fp32 GEMM: C[M,128] = A[M,128] @ W[128,128] ------
// grid.x = M/16 row tiles; block = 256 threads = 8 waves; wave w -> cols [16w,16w+16)
// A 16x4 fragment layout (32-bit A): lanes 0-15 hold M=lane, K={k,k+1};
//                                    lanes 16-31 hold M=lane-16, K={k+2,k+3}
// B 4x16 fragment layout (mirror):   lanes 0-15 hold N=lane, K={k,k+1};
//                                    lanes 16-31 hold N=lane-16, K={k+2,k+3}
// C/D: VGPR j, lanes 0-15 -> (M=j, N=lane); lanes 16-31 -> (M=j+8, N=lane-16)
__global__ __launch_bounds__(256) void gnn_gemm128_wmma(
    const float* __restrict__ A, const float* __restrict__ W,
    float* __restrict__ C, int M) {
  __shared__ float atile[16 * LDS_PITCH];

  const int tid  = threadIdx.x;
  const int wave = tid >> 5;
  const int lane = tid & 31;
  const int row0 = blockIdx.x * 16;
  const int col0 = wave * 16;

  // cooperative, coalesced stage of the 16x128 A tile into LDS
  for (int idx = tid; idx < 16 * F_DIM; idx += 256) {
    int r = idx >> 7;        // /128
    int c = idx & (F_DIM - 1);
    atile[r * LDS_PITCH + c] = A[(long long)(row0 + r) * F_DIM + c];
  }
  __syncthreads();

  const int m     = lane & 15;
  const int khalf = (lane >> 4) << 1;   // 0 for lanes 0-15, 2 for lanes 16-31
  const float* arow = &atile[m * LDS_PITCH + khalf];

  v8f acc = {};
#pragma unroll 4
  for (int k = 0; k < F_DIM; k += 4) {
    v2f a, b;
    a.x = arow[k];
    a.y = arow[k + 1];
    b.x = W[(long long)(k + khalf)     * F_DIM + col0 + m];
    b.y = W[(long long)(k + khalf + 1) * F_DIM + col0 + m];
    acc = __builtin_amdgcn_wmma_f32_16x16x4_f32(
        /*neg_a=*/false, a, /*neg_b=*/false, b,
        /*c_mod=*/(short)0, acc, /*reuse_a=*/false, /*reuse_b=*/false);
  }

  const int rbase = row0 + ((lane >> 4) << 3);  // +8 for upper half-lanes
  const int cc    = col0 + m;
#pragma unroll
  for (int j = 0; j < 8; ++j) {
    C[(long long)(rbase + j) * F_DIM + cc] = acc[j];
  }
}

// -------------------- edge scatter: one wave per edge -----------------------
// agg[col, :] += dinv[row]*ew*dinv[col] * h[row, :]
__global__ __launch_bounds__(256) void gnn_edge_scatter(
    const int* __restrict__ ei, const float* __restrict__ ew,
    const float* __restrict__ dinv, const float* __restrict__ h,
    float* __restrict__ agg, int nE) {
  int e = blockIdx.x * (blockDim.x >> 5) + (threadIdx.x >> 5);
  if (e >= nE) return;
  int lane = threadIdx.x & 31;
  int r = ei[e];
  int c = ei[(long long)nE + e];
  float norm = dinv[r] * ew[e] * dinv[c];
  const float4* hr = (const float4*)(h + (long long)r * F_DIM);
  float4 v = hr[lane];
  float* ac = agg + (long long)c * F_DIM + lane * 4;
  atomicAdd(ac + 0, norm * v.x);
  atomicAdd(ac + 1, norm * v.y);
  atomicAdd(ac + 2, norm * v.z);
  atomicAdd(ac + 3, norm * v.w);
}

// -------------------- fused self-loop + bias + relu (in place on agg) -------
__global__ void gnn_finish(float* __restrict__ agg, const float* __restrict__ h,
                           const float* __restrict__ dinv, const float* __restrict__ bias,
                           int n) {
  long long i = (long long)blockIdx.x * blockDim.x + threadIdx.x;
  if (i >= (long long)n * F_DIM) return;
  int node = (int)(i >> 7);
  int f = (int)(i & (F_DIM - 1));
  float d = dinv[node];
  float v = agg[i] + d * d * h[i] + bias[f];
  agg[i] = v > 0.0f ? v : 0.0f;
}

// -------------------- mean-pool accumulation: one wave per node -------------
__global__ __launch_bounds__(256) void gnn_pool(
    const float* __restrict__ h, const int* __restrict__ batch,
    float* __restrict__ sums, float* __restrict__ cnts, int n) {
  int node = blockIdx.x * (blockDim.x >> 5) + (threadIdx.x >> 5);
  if (node >= n) return;
  int lane = threadIdx.x & 31;
  int g = batch[node];
  const float4* hr = (const float4*)(h + (long long)node * F_DIM);
  float4 v = hr[lane];
  float* s = sums + (long long)g * F_DIM + lane * 4;
  atomicAdd(s + 0, v.x);
  atomicAdd(s + 1, v.y);
  atomicAdd(s + 2, v.z);
  atomicAdd(s + 3, v.w);
  if (lane == 0) atomicAdd(&cnts[g], 1.0f);
}

// -------------------- final: out[g] = (sums[g]/max(cnt,1)) . Wl + bl --------
__global__ __launch_bounds__(32) void gnn_final_linear(
    const float* __restrict__ sums, const float* __restrict__ cnts,
    const float* __restrict__ Wl, const float* __restrict__ bl,
    float* __restrict__ out) {
  int g = blockIdx.x;
  int lane = threadIdx.x;  // 32 lanes (wave32)
  float cnt = cnts[g];
  cnt = cnt > 1.0f ? cnt : 1.0f;
  float acc = 0.0f;
#pragma unroll
  for (int j = 0; j < F_DIM; j += 32) acc += sums[(long long)g * F_DIM + lane + j] * Wl[lane + j];
#pragma unroll
  for (int off = 16; off > 0; off >>= 1) acc += __shfl_xor(acc, off, 32);
  if (lane == 0) out[g] = acc / cnt + bl[0];
}

// ---------------------------------------------------------------------------
extern "C" void kernel_launch(void* const* d_in, const int* in_sizes, int n_in,
                              void* d_out, int out_size, void* d_ws, size_t ws_size,
                              hipStream_t stream) {
  const float* x    = (const float*)d_in[0];
  const int*   ei   = (const int*)d_in[1];   // [2, E]
  const int*   bat  = (const int*)d_in[2];   // [N]
  const float* ew   = (const float*)d_in[3]; // [E]
  const float* W1   = (const float*)d_in[4];
  const float* b1   = (const float*)d_in[5];
  const float* W2   = (const float*)d_in[6];
  const float* b2   = (const float*)d_in[7];
  const float* Wl   = (const float*)d_in[8];
  const float* bl   = (const float*)d_in[9];
  float* out = (float*)d_out;

  const int N  = in_sizes[0] / F_DIM;  // 100000
  const int NE = in_sizes[3];          // 1600000
  const int NG = out_size;             // 512

  // workspace layout
  char* ws = (char*)d_ws;
  float* dinv = (float*)(ws);                          // N floats (doubles as deg)
  float* sums = (float*)(ws + 401408);                 // NG*128 floats
  float* cnts = (float*)(ws + 401408 + 262144);        // NG floats
  float* bufA = (float*)(ws + (1 << 20));              // N*128 floats (h = x@W)
  float* bufB = bufA + (long long)N * F_DIM;           // N*128 floats (agg / relu'd h)

  const long long NF = (long long)N * F_DIM;
  const int fillBlocks = 2048;

  // ---- degree / normalization ----
  gnn_fill_f32<<<fillBlocks, 256, 0, stream>>>(dinv, 1.0f, N);  // self-loop weight
  gnn_deg_accum<<<(NE + 255) / 256, 256, 0, stream>>>(ei, ew, dinv, NE);
  gnn_rsqrt_inplace<<<(N + 255) / 256, 256, 0, stream>>>(dinv, N);

  const int gemmBlocks    = N / 16;                  // 6250 (N multiple of 16)
  const int scatterBlocks = (NE + 7) / 8;            // 8 edges (waves) per block
  const int poolBlocks    = (N + 7) / 8;
  const int finishBlocks  = (int)((NF + 255) / 256);

  // ---- layer 1 ----
  gnn_gemm128_wmma<<<gemmBlocks, 256, 0, stream>>>(x, W1, bufA, N);
  gnn_fill_f32<<<fillBlocks, 256, 0, stream>>>(bufB, 0.0f, NF);
  gnn_edge_scatter<<<scatterBlocks, 256, 0, stream>>>(ei, ew, dinv, bufA, bufB, NE);
  gnn_finish<<<finishBlocks, 256, 0, stream>>>(bufB, bufA, dinv, b1, N);

  // ---- layer 2 ----
  gnn_gemm128_wmma<<<gemmBlocks, 256, 0, stream>>>(bufB, W2, bufA, N);
  gnn_fill_f32<<<fillBlocks, 256, 0, stream>>>(bufB, 0.0f, NF);
  gnn_edge_scatter<<<scatterBlocks, 256, 0, stream>>>(ei, ew, dinv, bufA, bufB, NE);
  gnn_finish<<<finishBlocks, 256, 0, stream>>>(bufB, bufA, dinv, b2, N);

  // ---- mean pool + linear head ----
  gnn_fill_f32<<<64, 256, 0, stream>>>(sums, 0.0f, (long long)NG * F_DIM);
  gnn_fill_f32<<<4, 256, 0, stream>>>(cnts, 0.0f, NG);
  gnn_pool<<<poolBlocks, 256, 0, stream>>>(bufB, bat, sums, cnts, N);
  gnn_final_linear<<<NG, 32, 0, stream>>>(sums, cnts, Wl, bl, out);
}